// forward_warp_2911987827424
// MI455X (gfx1250) — compile-verified
//
#include <hip/hip_runtime.h>
#include <stdint.h>

// Problem constants per reference: B=4, C=3, H=W=1024
constexpr uint32_t Bn  = 4;
constexpr uint32_t Cn  = 3;
constexpr uint32_t Hn  = 1024;
constexpr uint32_t Wn  = 1024;
constexpr uint32_t HW  = Hn * Wn;          // 2^20
constexpr uint32_t CHW = Cn * HW;
constexpr uint32_t NPIX = Bn * HW;         // 4M pixels
constexpr uint32_t NIMG = Bn * CHW;        // 12M floats

__device__ __forceinline__ void atomic_fadd(float* p, float v) {
#if defined(__HIP_DEVICE_COMPILE__) && defined(__AMDGCN__)
    // HIP's documented route to the native global_atomic_add_f32 (no CAS expansion).
    (void)unsafeAtomicAdd(p, v);
#else
    (void)__hip_atomic_fetch_add(p, v, __ATOMIC_RELAXED, __HIP_MEMORY_SCOPE_AGENT);
#endif
}

// ---------------------------------------------------------------- forward splat
// FIRST in file so the disasm snippet shows the atomic lowering.
__global__ __launch_bounds__(256) void splat_kernel(const float*  __restrict__ im0,
                                                    const float*  __restrict__ flow,
                                                    float*        __restrict__ acc,
                                                    float*        __restrict__ wacc) {
    uint32_t tid = blockIdx.x * blockDim.x + threadIdx.x;
    if (tid >= NPIX) return;
    uint32_t b   = tid >> 20;
    uint32_t rem = tid & (HW - 1);
    uint32_t h   = rem >> 10;
    uint32_t w   = rem & (Wn - 1);

    const float2 f = ((const float2*)flow)[tid];
    __builtin_prefetch(&flow[2u * tid + 256u], 0, 1);  // global_prefetch_b8

    float x  = (float)w + f.x;
    float y  = (float)h + f.y;
    float x0 = floorf(x);
    float y0 = floorf(y);

    const float* plane = im0 + (size_t)b * CHW + rem;
    float v0 = plane[0];
    float v1 = plane[HW];
    float v2 = plane[2u * HW];

    float* accb  = acc  + (size_t)b * CHW;
    float* waccb = wacc + (size_t)b * HW;

#pragma unroll
    for (int corner = 0; corner < 4; ++corner) {
        float xi = x0 + (float)(corner & 1);
        float yi = y0 + (float)(corner >> 1);
        if (xi < 0.f || xi > (float)(Wn - 1) || yi < 0.f || yi > (float)(Hn - 1)) continue;
        float wt = (1.f - fabsf(x - xi)) * (1.f - fabsf(y - yi));
        uint32_t q = (uint32_t)yi * Wn + (uint32_t)xi;
        atomic_fadd(&accb[q],           v0 * wt);
        atomic_fadd(&accb[HW + q],      v1 * wt);
        atomic_fadd(&accb[2u * HW + q], v2 * wt);
        atomic_fadd(&waccb[q],          wt);
    }
}

// ---------------------------------------------------------------- zero init
__global__ __launch_bounds__(256) void zero_kernel(float4* __restrict__ img4,
                                                   float4* __restrict__ wacc4) {
    uint32_t tid = blockIdx.x * blockDim.x + threadIdx.x;
    const float4 z = make_float4(0.f, 0.f, 0.f, 0.f);
    if (tid < NIMG / 4) img4[tid] = z;
    if (tid < NPIX / 4) wacc4[tid] = z;
}

// --------------------------------------------- infill iteration 1 (in place, race-free)
// Holes hold exactly 0 after the splat, so pre-update corner values are
// (wacc[q]==0 ? 0 : img[q]); hole threads never read hole pixels -> no race.
__global__ __launch_bounds__(256) void infill1_kernel(const float* __restrict__ flowback,
                                                      const float* __restrict__ wacc,
                                                      float*       __restrict__ img,
                                                      const int*   __restrict__ niter) {
    if (*niter < 1) return;                       // uniform scalar load
    uint32_t tid = blockIdx.x * blockDim.x + threadIdx.x;
    if (tid >= NPIX) return;
    if (wacc[tid] != 0.f) return;                 // not a hole
    uint32_t b   = tid >> 20;
    uint32_t rem = tid & (HW - 1);
    uint32_t h   = rem >> 10;
    uint32_t w   = rem & (Wn - 1);

    const float2 f = ((const float2*)flowback)[tid];
    float x  = (float)w + f.x;
    float y  = (float)h + f.y;
    float x0 = floorf(x);
    float y0 = floorf(y);

    const float* imgb  = img  + (size_t)b * CHW;
    const float* waccb = wacc + (size_t)b * HW;
    float s0 = 0.f, s1 = 0.f, s2 = 0.f;

#pragma unroll
    for (int corner = 0; corner < 4; ++corner) {
        float xi = x0 + (float)(corner & 1);
        float yi = y0 + (float)(corner >> 1);
        if (xi < 0.f || xi > (float)(Wn - 1) || yi < 0.f || yi > (float)(Hn - 1)) continue;
        float wt = (1.f - fabsf(x - xi)) * (1.f - fabsf(y - yi));
        uint32_t q = (uint32_t)yi * Wn + (uint32_t)xi;
        if (waccb[q] != 0.f) {                    // stable non-hole pixel
            s0 += imgb[q]           * wt;
            s1 += imgb[HW + q]      * wt;
            s2 += imgb[2u * HW + q] * wt;
        }
    }
    float* out = img + (size_t)b * CHW + rem;
    out[0]       = s0;
    out[HW]      = s1;
    out[2u * HW] = s2;
}

// --------------------------------------------- infill iterations k >= 2 (ping-pong)
__global__ __launch_bounds__(256) void infillk_kernel(const float* __restrict__ src,
                                                      float*       __restrict__ dst,
                                                      const float* __restrict__ flowback,
                                                      const float* __restrict__ wacc,
                                                      const int*   __restrict__ niter,
                                                      int k) {
    if (*niter < k) return;                       // inactive iteration: near-free
    uint32_t tid = blockIdx.x * blockDim.x + threadIdx.x;
    if (tid >= NPIX) return;
    uint32_t b   = tid >> 20;
    uint32_t rem = tid & (HW - 1);

    const float* srcb = src + (size_t)b * CHW;
    float*       dstp = dst + (size_t)b * CHW + rem;

    if (wacc[tid] != 0.f) {                       // non-hole: copy through
        dstp[0]       = srcb[rem];
        dstp[HW]      = srcb[HW + rem];
        dstp[2u * HW] = srcb[2u * HW + rem];
        return;
    }

    uint32_t h = rem >> 10;
    uint32_t w = rem & (Wn - 1);
    const float2 f = ((const float2*)flowback)[tid];
    float x  = (float)w + f.x;
    float y  = (float)h + f.y;
    float x0 = floorf(x);
    float y0 = floorf(y);
    float s0 = 0.f, s1 = 0.f, s2 = 0.f;

#pragma unroll
    for (int corner = 0; corner < 4; ++corner) {
        float xi = x0 + (float)(corner & 1);
        float yi = y0 + (float)(corner >> 1);
        if (xi < 0.f || xi > (float)(Wn - 1) || yi < 0.f || yi > (float)(Hn - 1)) continue;
        float wt = (1.f - fabsf(x - xi)) * (1.f - fabsf(y - yi));
        uint32_t q = (uint32_t)yi * Wn + (uint32_t)xi;
        s0 += srcb[q]           * wt;             // k>=2: holes already filled in src
        s1 += srcb[HW + q]      * wt;
        s2 += srcb[2u * HW + q] * wt;
    }
    dstp[0]       = s0;
    dstp[HW]      = s1;
    dstp[2u * HW] = s2;
}

// --------------------------------------------- finalize: move result to d_out if parity ended in ws
__global__ __launch_bounds__(256) void finalize_kernel(const float4* __restrict__ A,
                                                       float4*       __restrict__ O,
                                                       const int*    __restrict__ niter) {
    int n = *niter;
    int m = n > 4 ? 4 : n;                        // executed iterations
    if (m != 2 && m != 4) return;                 // data already in d_out
    uint32_t tid = blockIdx.x * blockDim.x + threadIdx.x;
    if (tid < NIMG / 4) O[tid] = A[tid];
}

extern "C" void kernel_launch(void* const* d_in, const int* in_sizes, int n_in,
                              void* d_out, int out_size, void* d_ws, size_t ws_size,
                              hipStream_t stream) {
    const float* im0      = (const float*)d_in[0];
    const float* flow     = (const float*)d_in[1];
    const float* flowback = (const float*)d_in[2];
    const int*   niter    = (const int*)d_in[3];

    float* O    = (float*)d_out;                       // image buffer (B,C,H,W)
    float* A    = (float*)d_ws;                        // ping-pong image buffer (48 MB)
    float* wacc = (float*)d_ws + (size_t)NIMG;         // weight accumulator (16 MB)

    dim3 blk(256);
    dim3 grdZ((NIMG / 4 + 255) / 256);                 // covers both zero regions
    dim3 grdP((NPIX + 255) / 256);
    dim3 grdF((NIMG / 4 + 255) / 256);

    zero_kernel<<<grdZ, blk, 0, stream>>>((float4*)O, (float4*)wacc);
    splat_kernel<<<grdP, blk, 0, stream>>>(im0, flow, O, wacc);
    infill1_kernel<<<grdP, blk, 0, stream>>>(flowback, wacc, O, niter);
    infillk_kernel<<<grdP, blk, 0, stream>>>(O, A, flowback, wacc, niter, 2);
    infillk_kernel<<<grdP, blk, 0, stream>>>(A, O, flowback, wacc, niter, 3);
    infillk_kernel<<<grdP, blk, 0, stream>>>(O, A, flowback, wacc, niter, 4);
    finalize_kernel<<<grdF, blk, 0, stream>>>((const float4*)A, (float4*)O, niter);
}